// StaticPCALoss_63505386438787
// MI455X (gfx1250) — compile-verified
//
#include <hip/hip_runtime.h>
#include <hip/hip_bf16.h>

typedef __attribute__((ext_vector_type(2))) float v2f;
typedef __attribute__((ext_vector_type(8))) float v8f;

#define N_ROWS   1048576   // B*H*W = 256*64*64
#define HW       4096      // H*W
#define CHW      65536     // C*H*W
#define N_TILES  65536     // N_ROWS / 16
#define BLOCKS   2048
#define THREADS  256
#define WAVES    (BLOCKS * (THREADS / 32))
#define TILES_PER_WAVE (N_TILES / WAVES)   // = 4 exactly

__global__ __launch_bounds__(32) void pca_zero_ws(float* ws) {
    if (threadIdx.x < 16) ws[threadIdx.x] = 0.0f;
}

__global__ __launch_bounds__(THREADS) void pca_loss_main(
    const float* __restrict__ x,
    const float* __restrict__ xh,
    const float* __restrict__ pc,
    float* __restrict__ ws)
{
    __shared__ float chsum[16];
    const int lane   = threadIdx.x & 31;
    const int wib    = threadIdx.x >> 5;
    const int gwave  = blockIdx.x * (THREADS / 32) + wib;

    if (threadIdx.x < 16) chsum[threadIdx.x] = 0.0f;
    __syncthreads();

    const int m     = lane & 15;            // row-in-tile (A) / component k (D)
    const int cHalf = (lane >> 4) << 1;     // 0 for lanes 0-15, 2 for lanes 16-31

    // B fragments: 4x16 slices of PC (rows c..c+3, all 16 cols), resident in regs.
    // VGPR0: lanes0-15 row c+0, lanes16-31 row c+2; VGPR1: rows c+1 / c+3.
    v2f bfrag[4];
#pragma unroll
    for (int i = 0; i < 4; ++i) {
        const int c0 = 4 * i + cHalf;
        bfrag[i].x = pc[c0 * 16 + m];
        bfrag[i].y = pc[(c0 + 1) * 16 + m];
    }

    float acc = 0.0f;

#pragma unroll
    for (int t = 0; t < TILES_PER_WAVE; ++t) {
        const int tile = gwave * TILES_PER_WAVE + t;
        const int n0   = tile << 4;                 // first row of tile
        const int b    = n0 >> 12;                  // n0 / HW
        const int s    = (n0 & (HW - 1)) + m;       // spatial index for this lane's row
        const float* __restrict__ xb  = x  + b * CHW + s;
        const float* __restrict__ xhb = xh + b * CHW + s;

        v8f d = {0.f, 0.f, 0.f, 0.f, 0.f, 0.f, 0.f, 0.f};

#pragma unroll
        for (int i = 0; i < 4; ++i) {
            const int c0 = 4 * i + cHalf;
            v2f a;
            // streamed once: non-temporal loads, keep L2 clean
            a.x = __builtin_nontemporal_load(xb  + c0 * HW)
                - __builtin_nontemporal_load(xhb + c0 * HW);
            a.y = __builtin_nontemporal_load(xb  + (c0 + 1) * HW)
                - __builtin_nontemporal_load(xhb + (c0 + 1) * HW);
            // D(16x16,f32) += A(16x4,f32) * B(4x16,f32), chained over K
            d = __builtin_amdgcn_wmma_f32_16x16x4_f32(
                    false, a, false, bfrag[i], (short)0, d, false, false);
        }

        // lane L holds coeff[rows, k=L%16] in 8 VGPRs -> per-component sq-sum
#pragma unroll
        for (int v = 0; v < 8; ++v) acc = fmaf(d[v], d[v], acc);
    }

    // merge the two half-wave copies of component k = lane%16
    acc += __shfl_xor(acc, 16, 32);
    if (lane < 16) atomicAdd(&chsum[m], acc);   // ds_add_f32
    __syncthreads();
    if (threadIdx.x < 16) unsafeAtomicAdd(&ws[threadIdx.x], chsum[threadIdx.x]);
}

__global__ __launch_bounds__(32) void pca_finalize(const float* __restrict__ ws,
                                                   float* __restrict__ out)
{
    const float invN = 1.0f / (float)N_ROWS;
    const int t = threadIdx.x;
    if (t < 16) out[1 + t] = ws[t] * invN;          // channel_losses
    if (t == 0) {
        float s = 0.0f;
#pragma unroll
        for (int i = 0; i < 16; ++i) s += ws[i];
        out[0] = s * invN * (1.0f / 16.0f);         // total_loss
    }
}

extern "C" void kernel_launch(void* const* d_in, const int* in_sizes, int n_in,
                              void* d_out, int out_size, void* d_ws, size_t ws_size,
                              hipStream_t stream) {
    const float* x   = (const float*)d_in[0];
    const float* xh  = (const float*)d_in[1];
    // d_in[2] is `mean` — it cancels algebraically: (x-mu)P - (xh-mu)P = (x-xh)P
    const float* pc  = (const float*)d_in[3];
    float*       out = (float*)d_out;
    float*       ws  = (float*)d_ws;

    pca_zero_ws<<<1, 32, 0, stream>>>(ws);
    pca_loss_main<<<BLOCKS, THREADS, 0, stream>>>(x, xh, pc, ws);
    pca_finalize<<<1, 32, 0, stream>>>(ws, out);
}